// SeedModel_EdgeWeightedGAT_3504693313918
// MI455X (gfx1250) — compile-verified
//
#include <hip/hip_runtime.h>
#include <hip/hip_bf16.h>

// Edge-weighted 2-branch GAT (3 layers) + MLP head for MI455X (gfx1250).
// GEMMs via V_WMMA_F32_16X16X4_F32 (fp32-exact); edge softmax/scatter via
// per-node score factorization + f32 global atomics (L2-resident buffers).

#define NN 50000
#define NE 800000
#define FIN 128
#define HID 64
#define NEG_SLOPE 0.2f
#define BN_EPS 1e-5f

typedef __attribute__((ext_vector_type(2))) float v2f;
typedef __attribute__((ext_vector_type(8))) float v8f;

__global__ void k_zero(float* __restrict__ p, int n) {
  int i = blockIdx.x * blockDim.x + threadIdx.x;
  if (i < n) p[i] = 0.0f;
}

// Y[N,64] = X[N,K] @ W[64,K]^T (+bias)(+relu).  N multiple of 16, K multiple of 4.
// One wave per 16x16 output tile; wave-uniform control flow (EXEC all ones).
__global__ void k_gemm_wmma(const float* __restrict__ X, const float* __restrict__ W,
                            float* __restrict__ Y, int N, int K,
                            const float* __restrict__ bias, int do_relu) {
  int wave = (blockIdx.x * blockDim.x + threadIdx.x) >> 5;
  int lane = threadIdx.x & 31;
  int ntiles = (N >> 4) * (HID >> 4);
  if (wave >= ntiles) return;
  int tile_m = wave & 3;   // feature block (HID/16 == 4)
  int tile_n = wave >> 2;  // node block
  int node0 = tile_n << 4;
  int feat0 = tile_m << 4;
  int r = lane & 15;
  int half = lane >> 4;
  // A: 16x4 f32, lanes 0-15 -> k0,k0+1 ; lanes 16-31 -> k0+2,k0+3
  const float* xrow = X + (size_t)(node0 + r) * K + half * 2;
  // B: 4x16 f32 = W^T tile; same per-lane pattern with row = output feature
  const float* wrow = W + (size_t)(feat0 + r) * K + half * 2;
  v8f c = {};
  for (int k = 0; k < K; k += 4) {
    v2f a = *(const v2f*)(xrow + k);
    v2f b = *(const v2f*)(wrow + k);
    c = __builtin_amdgcn_wmma_f32_16x16x4_f32(false, a, false, b, (short)0, c,
                                              false, false);
  }
  int col = feat0 + r;
  float bv = bias ? bias[col] : 0.0f;
#pragma unroll
  for (int i = 0; i < 8; ++i) {
    int row = node0 + i + half * 8;  // C/D: vgpr i -> M=i (lanes 0-15) / M=i+8
    float v = c[i] + bv;
    if (do_relu) v = fmaxf(v, 0.0f);
    Y[(size_t)row * HID + col] = v;
  }
}

// Per-node attention scores: s_dst[n] = h[n]·att[0:64], s_src[n] = h[n]·att[64:128]
__global__ void k_scores(const float* __restrict__ h, const float* __restrict__ att,
                         float* __restrict__ s_dst, float* __restrict__ s_src, int N) {
  int wid = (blockIdx.x * blockDim.x + threadIdx.x) >> 5;
  int lane = threadIdx.x & 31;
  if (wid >= N) return;
  const float* hp = h + (size_t)wid * HID;
  float h0 = hp[lane], h1 = hp[lane + 32];
  float ai = h0 * att[lane] + h1 * att[lane + 32];
  float aj = h0 * att[HID + lane] + h1 * att[HID + lane + 32];
  for (int o = 16; o > 0; o >>= 1) {
    ai += __shfl_xor(ai, o, 32);
    aj += __shfl_xor(aj, o, 32);
  }
  if (lane == 0) { s_dst[wid] = ai; s_src[wid] = aj; }
}

__device__ __forceinline__ void edge_sd(const int* __restrict__ ei, int e, int E,
                                        int& s, int& d) {
  if (e < E) { s = ei[e]; d = ei[E + e]; }
  else { s = e - E; d = e - E; }  // appended self loops
}

__device__ __forceinline__ float leaky(float a) {
  return a >= 0.0f ? a : NEG_SLOPE * a;
}

// m[d] = max over edges of positive alpha (m pre-zeroed => max(0, max alpha))
__global__ void k_edge_max(const int* __restrict__ ei, const float* __restrict__ s_dst,
                           const float* __restrict__ s_src, unsigned int* __restrict__ mbits,
                           int E, int total) {
  int e = blockIdx.x * blockDim.x + threadIdx.x;
  if (e >= total) return;
  int s, d; edge_sd(ei, e, E, s, d);
  float a = leaky(s_dst[d] + s_src[s]);
  if (a > 0.0f) atomicMax(&mbits[d], __float_as_uint(a));  // positives order as uints
}

__global__ void k_edge_sum(const int* __restrict__ ei, const float* __restrict__ s_dst,
                           const float* __restrict__ s_src, const float* __restrict__ m,
                           float* __restrict__ denom, int E, int total) {
  int e = blockIdx.x * blockDim.x + threadIdx.x;
  if (e >= total) return;
  int s, d; edge_sd(ei, e, E, s, d);
  float a = leaky(s_dst[d] + s_src[s]);
  atomicAdd(&denom[d], __expf(a - m[d]));
}

__global__ void k_edge_coef(const int* __restrict__ ei, const float* __restrict__ ew,
                            const float* __restrict__ s_dst, const float* __restrict__ s_src,
                            const float* __restrict__ m, const float* __restrict__ denom,
                            float* __restrict__ coef, int E, int total) {
  int e = blockIdx.x * blockDim.x + threadIdx.x;
  if (e >= total) return;
  int s, d; edge_sd(ei, e, E, s, d);
  float a = leaky(s_dst[d] + s_src[s]);
  float ex = __expf(a - m[d]);
  float w = (e < E) ? ew[e] : 1.0f;
  coef[e] = ex / (denom[d] + 1e-16f) * w;
}

// y[d,:] += coef[e] * h[s,:]  — 64 threads per edge, coalesced per feature
__global__ void k_scatter(const int* __restrict__ ei, const float* __restrict__ coef,
                          const float* __restrict__ h, float* __restrict__ y,
                          int E, int total) {
  long long idx = (long long)blockIdx.x * blockDim.x + threadIdx.x;
  if (idx >= (long long)total * HID) return;
  int e = (int)(idx >> 6);
  int c = (int)(idx & 63);
  int s, d; edge_sd(ei, e, E, s, d);
  atomicAdd(&y[(size_t)d * HID + c], coef[e] * h[(size_t)s * HID + c]);
}

// per-feature sum / sumsq partials (256 nodes per block -> 128 atomics/block)
__global__ void k_bn_stats(const float* __restrict__ y, float* __restrict__ gsum,
                           float* __restrict__ gsq, int N) {
  __shared__ float ssum[256], ssq[256];
  int c = threadIdx.x & 63, g = threadIdx.x >> 6;
  int n0 = blockIdx.x * 256;
  float sm = 0.0f, sq = 0.0f;
  int nend = n0 + 256 < N ? n0 + 256 : N;
  for (int n = n0 + g; n < nend; n += 4) {
    float v = y[(size_t)n * HID + c];
    sm += v; sq += v * v;
  }
  ssum[threadIdx.x] = sm; ssq[threadIdx.x] = sq;
  __syncthreads();
  if (g == 0) {
    float ts = ssum[c] + ssum[64 + c] + ssum[128 + c] + ssum[192 + c];
    float tq = ssq[c] + ssq[64 + c] + ssq[128 + c] + ssq[192 + c];
    atomicAdd(&gsum[c], ts);
    atomicAdd(&gsq[c], tq);
  }
}

__global__ void k_bn_final(const float* __restrict__ gsum, const float* __restrict__ gsq,
                           const float* __restrict__ gamma, const float* __restrict__ beta,
                           float* __restrict__ scale, float* __restrict__ shift, int N) {
  int c = threadIdx.x;
  if (c >= HID) return;
  float mu = gsum[c] / (float)N;
  float var = gsq[c] / (float)N - mu * mu;  // biased var == jnp.var
  float a = gamma[c] * rsqrtf(var + BN_EPS);
  scale[c] = a;
  shift[c] = beta[c] - mu * a;
}

// relu(bn(y)) -> dst with arbitrary row stride / column offset
__global__ void k_bn_apply(const float* __restrict__ y, const float* __restrict__ scale,
                           const float* __restrict__ shift, float* __restrict__ dst,
                           int stride, int coloff, int N) {
  int idx = blockIdx.x * blockDim.x + threadIdx.x;
  if (idx >= N * HID) return;
  int n = idx >> 6, c = idx & 63;
  float v = fmaxf(y[idx] * scale[c] + shift[c], 0.0f);
  dst[(size_t)n * stride + coloff + c] = v;
}

// out[n] = z[n]·w2 + b2   (one wave per node)
__global__ void k_predict_out(const float* __restrict__ z, const float* __restrict__ w2,
                              const float* __restrict__ b2, float* __restrict__ out, int N) {
  int wid = (blockIdx.x * blockDim.x + threadIdx.x) >> 5;
  int lane = threadIdx.x & 31;
  if (wid >= N) return;
  const float* zp = z + (size_t)wid * HID;
  float acc = zp[lane] * w2[lane] + zp[lane + 32] * w2[lane + 32];
  for (int o = 16; o > 0; o >>= 1) acc += __shfl_xor(acc, o, 32);
  if (lane == 0) out[wid] = acc + b2[0];
}

extern "C" void kernel_launch(void* const* d_in, const int* in_sizes, int n_in,
                              void* d_out, int out_size, void* d_ws, size_t ws_size,
                              hipStream_t stream) {
  (void)in_sizes; (void)n_in; (void)out_size; (void)ws_size;
  const int N = NN, E = NE, T = NE + NN;

  // d_in layout (setup_inputs insertion order; params leaves in insertion order):
  // 0:x 1:ei0 2:ei1 3:ew0 4:ew1
  // 5..10  conv0: (W,att) x3   11..16 bn0: (g,b) x3
  // 17..22 conv1: (W,att) x3   23..28 bn1: (g,b) x3
  // 29:pred.W1 30:pred.b1 31:pred.W2 32:pred.b2
  const float* x = (const float*)d_in[0];

  float* ws = (float*)d_ws;
  size_t off = 0;
  auto carve = [&](size_t n) {
    float* p = ws + off;
    off += (n + 63) & ~(size_t)63;
    return p;
  };
  float* bufH = carve((size_t)N * HID);       // gemm output / MLP hidden
  float* bufY = carve((size_t)N * HID);       // message accumulator
  float* bufX = carve((size_t)N * HID);       // next-layer input
  float* xcat = carve((size_t)N * 2 * HID);   // branch concat
  float* s_dst = carve(N);
  float* s_src = carve(N);
  float* coef  = carve(T);
  float* mmax  = carve(N);
  float* gden  = carve(N);
  float* gsum  = carve(HID);                  // contiguous with gsq (64-aligned carve)
  float* gsq   = carve(HID);
  float* scale = carve(HID);
  float* shift = carve(HID);

  const int gemm_waves  = (N / 16) * (HID / 16);     // 12500 tiles
  const int gemm_blocks = (gemm_waves + 7) / 8;      // 8 waves / 256-thread block
  const int nodeW_blocks = (N + 7) / 8;              // one wave per node
  const int edge_blocks = (T + 255) / 256;
  const int nh_blocks   = (N * HID + 255) / 256;
  const int scat_blocks = (int)(((long long)T * HID + 255) / 256);

  for (int br = 0; br < 2; ++br) {
    const int*   ei = (const int*)d_in[1 + br];
    const float* ew = (const float*)d_in[3 + br];
    const float* xin = x;
    int K = FIN;
    int pbase = 5 + br * 12;
    for (int l = 0; l < 3; ++l) {
      const float* W   = (const float*)d_in[pbase + 2 * l];
      const float* att = (const float*)d_in[pbase + 1 + 2 * l];
      const float* gam = (const float*)d_in[pbase + 6 + 2 * l];
      const float* bet = (const float*)d_in[pbase + 7 + 2 * l];

      k_gemm_wmma<<<gemm_blocks, 256, 0, stream>>>(xin, W, bufH, N, K, nullptr, 0);
      k_scores<<<nodeW_blocks, 256, 0, stream>>>(bufH, att, s_dst, s_src, N);

      k_zero<<<nh_blocks, 256, 0, stream>>>(bufY, N * HID);
      k_zero<<<(N + 255) / 256, 256, 0, stream>>>(mmax, N);
      k_zero<<<(N + 255) / 256, 256, 0, stream>>>(gden, N);
      k_zero<<<1, 256, 0, stream>>>(gsum, 2 * HID);  // gsum+gsq contiguous

      k_edge_max<<<edge_blocks, 256, 0, stream>>>(ei, s_dst, s_src,
                                                  (unsigned int*)mmax, E, T);
      k_edge_sum<<<edge_blocks, 256, 0, stream>>>(ei, s_dst, s_src, mmax, gden, E, T);
      k_edge_coef<<<edge_blocks, 256, 0, stream>>>(ei, ew, s_dst, s_src, mmax, gden,
                                                   coef, E, T);
      k_scatter<<<scat_blocks, 256, 0, stream>>>(ei, coef, bufH, bufY, E, T);

      k_bn_stats<<<(N + 255) / 256, 256, 0, stream>>>(bufY, gsum, gsq, N);
      k_bn_final<<<1, 64, 0, stream>>>(gsum, gsq, gam, bet, scale, shift, N);
      if (l < 2)
        k_bn_apply<<<nh_blocks, 256, 0, stream>>>(bufY, scale, shift, bufX, HID, 0, N);
      else
        k_bn_apply<<<nh_blocks, 256, 0, stream>>>(bufY, scale, shift, xcat, 2 * HID,
                                                  br * HID, N);
      xin = bufX;
      K = HID;
    }
  }

  const float* W1 = (const float*)d_in[29];
  const float* b1 = (const float*)d_in[30];
  const float* W2 = (const float*)d_in[31];
  const float* b2 = (const float*)d_in[32];
  k_gemm_wmma<<<gemm_blocks, 256, 0, stream>>>(xcat, W1, bufH, N, 2 * HID, b1, 1);
  k_predict_out<<<nodeW_blocks, 256, 0, stream>>>(bufH, W2, b2, (float*)d_out, N);
}